// Decoder_75557064671267
// MI455X (gfx1250) — compile-verified
//
#include <hip/hip_runtime.h>
#include <cmath>

typedef float v2f __attribute__((ext_vector_type(2)));
typedef float v8f __attribute__((ext_vector_type(8)));

#define B_   128
#define T_   24
#define E_   512
#define H_   512
#define V_   12000
#define G3H  1536

// D = A(16x4 f32) * B(4x16 f32) + C(16x16 f32), wave32, fp32 FMA (RNE)
__device__ __forceinline__ v8f wmma_f32(v2f a, v2f b, v8f c) {
  return __builtin_amdgcn_wmma_f32_16x16x4_f32(
      /*neg_a=*/false, a, /*neg_b=*/false, b,
      /*c_mod=*/(short)0, c, /*reuse_a=*/false, /*reuse_b=*/false);
}

// ---------------------------------------------------------------------------
// x_cat[b,t,:] = (t==0) ? img[b,:] : emb[cap[b,t-1],:]
// grid = B*T rows, block = 128 threads, float4 copies (E=512 -> 128 float4)
// ---------------------------------------------------------------------------
__global__ __launch_bounds__(128) void gather_xcat(
    const float* __restrict__ img, const int* __restrict__ cap,
    const float* __restrict__ emb, float* __restrict__ xcat) {
  const int row = blockIdx.x;          // b*T + t
  const int b = row / T_;
  const int t = row % T_;
  const float4* src;
  if (t == 0) src = (const float4*)(img + (size_t)b * E_);
  else        src = (const float4*)(emb + (size_t)cap[b * T_ + (t - 1)] * E_);
  float4* dst = (float4*)(xcat + (size_t)row * E_);
  dst[threadIdx.x] = src[threadIdx.x];
}

// ---------------------------------------------------------------------------
// out[m,n] = sum_k X[m,k]*W[n,k] + bias[n]   (K = 512, X/W row-major)
// block = 8 waves (4 along M x 2 along N); each wave computes a 48x48 tile
// (3x3 = 9 f32 accumulators).  Per K-chunk of 4: 6 float2 loads, 9 WMMAs
// (1.5 WMMA per VMEM op).  Block tile = 192 x 96.
// ---------------------------------------------------------------------------
template <int NOUT>
__global__ __launch_bounds__(256) void gemm_nt_bias(
    const float* __restrict__ X, const float* __restrict__ W,
    const float* __restrict__ bias, float* __restrict__ out) {
  const int lane = threadIdx.x & 31;
  const int w    = threadIdx.x >> 5;
  const int half = lane >> 4;          // K phase: lanes 16-31 hold K=2,3
  const int lr   = lane & 15;          // A: row index, B: col index
  const int m0 = blockIdx.y * 192 + (w >> 1) * 48;
  const int n0 = blockIdx.x * 96  + (w & 1)  * 48;

  const float* xr0 = X + (size_t)(m0 + lr) * 512 + 2 * half;
  const float* xr1 = xr0 + (size_t)16 * 512;
  const float* xr2 = xr0 + (size_t)32 * 512;
  const float* wp0 = W + (size_t)(n0 + lr) * 512 + 2 * half;
  const float* wp1 = wp0 + (size_t)16 * 512;
  const float* wp2 = wp0 + (size_t)32 * 512;

  v8f c00 = {}, c01 = {}, c02 = {};
  v8f c10 = {}, c11 = {}, c12 = {};
  v8f c20 = {}, c21 = {}, c22 = {};
#pragma unroll 2
  for (int k = 0; k < 512; k += 4) {
    v2f a0 = *(const v2f*)(xr0 + k);
    v2f a1 = *(const v2f*)(xr1 + k);
    v2f a2 = *(const v2f*)(xr2 + k);
    v2f b0 = *(const v2f*)(wp0 + k);
    v2f b1 = *(const v2f*)(wp1 + k);
    v2f b2 = *(const v2f*)(wp2 + k);
    c00 = wmma_f32(a0, b0, c00); c01 = wmma_f32(a0, b1, c01); c02 = wmma_f32(a0, b2, c02);
    c10 = wmma_f32(a1, b0, c10); c11 = wmma_f32(a1, b1, c11); c12 = wmma_f32(a1, b2, c12);
    c20 = wmma_f32(a2, b0, c20); c21 = wmma_f32(a2, b1, c21); c22 = wmma_f32(a2, b2, c22);
  }

  const float bb0 = bias[n0 + lr];
  const float bb1 = bias[n0 + 16 + lr];
  const float bb2 = bias[n0 + 32 + lr];
  // C/D layout: VGPR i -> row (tile_m0 + 8*half + i), col (tile_n0 + lr)
  float* o0 = out + (size_t)(m0 + 8 * half) * NOUT + n0 + lr;
  float* o1 = o0 + (size_t)16 * NOUT;
  float* o2 = o0 + (size_t)32 * NOUT;
#pragma unroll
  for (int i = 0; i < 8; ++i) {
    const size_t r = (size_t)i * NOUT;
    o0[r] = c00[i] + bb0;  o0[r + 16] = c01[i] + bb1;  o0[r + 32] = c02[i] + bb2;
    o1[r] = c10[i] + bb0;  o1[r + 16] = c11[i] + bb1;  o1[r + 32] = c12[i] + bb2;
    o2[r] = c20[i] + bb0;  o2[r + 16] = c21[i] + bb1;  o2[r + 32] = c22[i] + bb2;
  }
}

// ---------------------------------------------------------------------------
// One GRU time step. hs stored [B, T, H] (row b*T+t).
// Each wave: one 16x16 tile of h_new; 3 WMMA accumulators for r/z/n columns
// of gh = h_prev @ W_hh^T.  gh bias b_hh added here; gx already has b_ih.
// ---------------------------------------------------------------------------
__global__ __launch_bounds__(256) void gru_step(
    const float* __restrict__ gx, const float* __restrict__ W_hh,
    const float* __restrict__ b_hh, float* __restrict__ hs, int t) {
  const int lane = threadIdx.x & 31;
  const int w    = threadIdx.x >> 5;
  const int half = lane >> 4;
  const int lr   = lane & 15;
  const int tile = blockIdx.x * 8 + w;   // 0..255
  const int m0 = (tile >> 5) * 16;       // batch tile (8 of them)
  const int n0 = (tile & 31) * 16;       // hidden tile (32 of them)

  v8f cr = {}, cz = {}, cn = {};
  if (t > 0) {   // uniform branch: EXEC stays all-ones around WMMA
    const float* xrow = hs + ((size_t)(m0 + lr) * T_ + (t - 1)) * H_ + 2 * half;
    const float* wr = W_hh + (size_t)(n0 + lr) * H_ + 2 * half;
    const float* wz = wr + (size_t)H_ * H_;
    const float* wn = wr + (size_t)2 * H_ * H_;
#pragma unroll 4
    for (int k = 0; k < H_; k += 4) {
      v2f a  = *(const v2f*)(xrow + k);
      v2f br = *(const v2f*)(wr + k);
      v2f bz = *(const v2f*)(wz + k);
      v2f bn = *(const v2f*)(wn + k);
      cr = wmma_f32(a, br, cr);
      cz = wmma_f32(a, bz, cz);
      cn = wmma_f32(a, bn, cn);
    }
  }

  const int col = n0 + lr;
  const float bhr = b_hh[col];
  const float bhz = b_hh[H_ + col];
  const float bhn = b_hh[2 * H_ + col];
#pragma unroll
  for (int i = 0; i < 8; ++i) {
    const int b = m0 + 8 * half + i;
    const size_t gxb = ((size_t)b * T_ + t) * G3H;
    const float xr = gx[gxb + col];
    const float xz = gx[gxb + H_ + col];
    const float xn = gx[gxb + 2 * H_ + col];
    const float hr = cr[i] + bhr;
    const float hz = cz[i] + bhz;
    const float hn = cn[i] + bhn;
    const float r = 1.0f / (1.0f + __expf(-(xr + hr)));
    const float z = 1.0f / (1.0f + __expf(-(xz + hz)));
    const float n = tanhf(xn + r * hn);
    const float hp = (t > 0) ? hs[((size_t)b * T_ + (t - 1)) * H_ + col] : 0.0f;
    hs[((size_t)b * T_ + t) * H_ + col] = (1.0f - z) * n + z * hp;
  }
}

// ---------------------------------------------------------------------------
extern "C" void kernel_launch(void* const* d_in, const int* in_sizes, int n_in,
                              void* d_out, int out_size, void* d_ws, size_t ws_size,
                              hipStream_t stream) {
  const float* img   = (const float*)d_in[0];
  const int*   cap   = (const int*)  d_in[1];
  // d_in[2] = caplen (unused: static unroll length)
  const float* emb   = (const float*)d_in[3];
  const float* W_ih  = (const float*)d_in[4];
  const float* W_hh  = (const float*)d_in[5];
  const float* b_ih  = (const float*)d_in[6];
  const float* b_hh  = (const float*)d_in[7];
  const float* W_out = (const float*)d_in[8];
  const float* b_out = (const float*)d_in[9];
  float* out = (float*)d_out;

  // Workspace: x_cat [B,T,E] | gx [B,T,3H] | hs [B,T,H]  (~31.5 MB total)
  float* xcat = (float*)d_ws;
  float* gx   = xcat + (size_t)B_ * T_ * E_;
  float* hs   = gx   + (size_t)B_ * T_ * G3H;

  // 1) embedding gather + img concat
  gather_xcat<<<B_ * T_, 128, 0, stream>>>(img, cap, emb, xcat);
  // 2) gx = x_cat @ W_ih^T + b_ih   (M=3072, N=1536, K=512)
  gemm_nt_bias<G3H><<<dim3(16, 16), 256, 0, stream>>>(xcat, W_ih, b_ih, gx);
  // 3) sequential GRU recurrence (stream-ordered)
  for (int t = 0; t < T_; ++t)
    gru_step<<<32, 256, 0, stream>>>(gx, W_hh, b_hh, hs, t);
  // 4) logits = hs @ W_out^T + b_out   (M=3072, N=12000, K=512)
  gemm_nt_bias<V_><<<dim3(125, 16), 256, 0, stream>>>(hs, W_out, b_out, out);
}